// TemporalChannelIndependentLR_43679817401113
// MI455X (gfx1250) — compile-verified
//
#include <hip/hip_runtime.h>

typedef __attribute__((ext_vector_type(2))) float v2f;
typedef __attribute__((ext_vector_type(4))) float v4f;
typedef __attribute__((ext_vector_type(8))) float v8f;

#define Bsz   1024
#define Ssz   336
#define Csz   64
#define Osz   168
#define Tsz   96
#define KO    28          // K-chunk staged in LDS (row stride 28 words: bank-conflict-free)
#define NCHUNK 6          // 168 / 28
#define KSTEPS 7          // 28 / 4  (WMMA K=4 per step)
#define NT    6           // 96 / 16 N-tiles per wave

__global__ __launch_bounds__(256)
void tcilr_wmma_kernel(const float* __restrict__ inp,
                       const float* __restrict__ W,
                       const float* __restrict__ bias,
                       float* __restrict__ out)
{
    // Phase 1 (A stage): [c8][b16][o28] = 3584 floats.
    // Phase 2 (D stage): [m16][n96][c8] = 12288 floats (48 KB) -> sized for phase 2.
    __shared__ float smem[16 * Tsz * 8];

    const int tid  = threadIdx.x;
    const int lane = tid & 31;
    const int wave = tid >> 5;      // channel within group (0..7)
    const int half = lane >> 4;     // K pair {0,1} vs {2,3}
    const int row  = lane & 15;     // M row for A, N col for B/D

    const int mBase = blockIdx.x * 16;
    const int cBase = blockIdx.y * 8;
    const int c     = cBase + wave;

    v8f acc[NT];
    #pragma unroll
    for (int n = 0; n < NT; ++n)
        acc[n] = (v8f){0.f,0.f,0.f,0.f,0.f,0.f,0.f,0.f};

    // W base for this wave: fold per-lane row + K-half offset into one base,
    // so every inner load is base + compile-time immediate.
    const float* Wbase = W + (size_t)c * (Osz * Tsz) + row + half * 2 * Tsz;

    // A-stage mapping: 4 threads cover the 8 contiguous channels (float2 each)
    // -> 448 (b,o) pairs / 64 slots = 7 exact passes, no remainder branch.
    const int cc2   = (tid & 3) * 2;
    const int slot2 = tid >> 2;     // 0..63

    const float* As = &smem[wave * (16 * KO) + row * KO + half * 2];

    for (int ch = 0; ch < NCHUNK; ++ch) {
        const int ko = ch * KO;

        // ---- stage A chunk into LDS, layout [c][b][o] ----
        #pragma unroll
        for (int p = 0; p < 7; ++p) {
            int lin = p * 64 + slot2;          // 0..447
            int bb  = lin / KO;
            int oo  = lin - bb * KO;
            const float* g = inp + (size_t)(mBase + bb) * (Ssz * Csz)
                                 + (size_t)(Ssz - Osz + ko + oo) * Csz
                                 + cBase + cc2;
            v2f v = *(const v2f*)g;            // global_load_b64, coalesced along C
            int l0 = cc2 * (16 * KO) + bb * KO + oo;
            smem[l0           ] = v.x;         // -> ds_store_2addr_stride64_b32
            smem[l0 + 16 * KO ] = v.y;
        }
        __syncthreads();

        // ---- compute: 7 k-steps x 6 N-tiles of WMMA f32 16x16x4 ----
        const float* Wk = Wbase + (size_t)ko * Tsz;
        #pragma unroll
        for (int j = 0; j < KSTEPS; ++j) {
            v2f a = *(const v2f*)(As + j * 4); // ds_load_b64, conflict-free
            #pragma unroll
            for (int n = 0; n < NT; ++n) {
                v2f b;
                b.x = Wk[ j * 4      * Tsz + n * 16];   // imm-offset b32 loads
                b.y = Wk[(j * 4 + 1) * Tsz + n * 16];
                acc[n] = __builtin_amdgcn_wmma_f32_16x16x4_f32(
                             false, a, false, b, (short)0, acc[n], false, false);
            }
        }
        __syncthreads();
    }

    // ---- bias + stage all 6 D tiles to LDS as [m][n96][c8] ----
    #pragma unroll
    for (int n = 0; n < NT; ++n) {
        float bv = bias[c * Tsz + n * 16 + row];
        #pragma unroll
        for (int v = 0; v < 8; ++v) {
            int m = v + half * 8;
            smem[(m * Tsz + n * 16 + row) * 8 + wave] = acc[n][v] + bv;
        }
    }
    __syncthreads();

    // ---- cooperative float4 stores: out[b][t][c], c contiguous ----
    const int cc4  = (tid & 1) * 4;
    const int slot = tid >> 1;                 // 0..127; 1536 (m,n) pairs -> 12 passes
    #pragma unroll
    for (int p = 0; p < 12; ++p) {
        int lin = p * 128 + slot;
        int m = lin / Tsz;
        int n = lin - m * Tsz;
        v4f v = *(const v4f*)(&smem[(m * Tsz + n) * 8 + cc4]);   // ds_load_b128
        float* g = out + (size_t)(mBase + m) * (Tsz * Csz)
                       + (size_t)n * Csz + cBase + cc4;
        *(v4f*)g = v;                          // global_store_b128
    }
}

extern "C" void kernel_launch(void* const* d_in, const int* in_sizes, int n_in,
                              void* d_out, int out_size, void* d_ws, size_t ws_size,
                              hipStream_t stream) {
    const float* inp  = (const float*)d_in[0];   // [1024, 336, 64] fp32
    const float* W    = (const float*)d_in[1];   // [64, 168, 96]  fp32
    const float* bias = (const float*)d_in[2];   // [64, 96]       fp32
    float* out = (float*)d_out;                  // [1024, 96, 64] fp32

    dim3 grid(Bsz / 16, Csz / 8);                // (64, 8)
    tcilr_wmma_kernel<<<grid, 256, 0, stream>>>(inp, W, bias, out);
}